// ResidualGraphLayer_56143812493336
// MI455X (gfx1250) — compile-verified
//
#include <hip/hip_runtime.h>
#include <math.h>

// ---------------- constants from the reference ----------------
#define DIMC      256
#define HEADS     4
#define HEADDIM   64
#define NEG_SLOPE 0.2f
#define LN_EPS    1e-5f
#define KCHUNK    128          // K-chunk staged in LDS for GEMM B tiles

typedef __attribute__((ext_vector_type(2))) float v2f;
typedef __attribute__((ext_vector_type(8))) float v8f;
typedef __attribute__((ext_vector_type(4))) int   v4i;

// ---------------- CDNA5 async global->LDS staging (ASYNCcnt path) ----------------
#if defined(__gfx1250__) && defined(__HIP_DEVICE_COMPILE__) && \
    __has_builtin(__builtin_amdgcn_global_load_async_to_lds_b128) && \
    __has_builtin(__builtin_amdgcn_s_wait_asynccnt)
#define USE_ASYNC_LDS 1
#else
#define USE_ASYNC_LDS 0
#endif

__device__ __forceinline__ void stage16_to_lds(float* lds_dst, const float* gsrc) {
#if USE_ASYNC_LDS
    __builtin_amdgcn_global_load_async_to_lds_b128(
        (__attribute__((address_space(1))) v4i*)gsrc,
        (__attribute__((address_space(3))) v4i*)lds_dst,
        0, 0);
#else
    *(float4*)lds_dst = *(const float4*)gsrc;
#endif
}

__device__ __forceinline__ void stage_wait() {
#if USE_ASYNC_LDS
    __builtin_amdgcn_s_wait_asynccnt(0);
#endif
}

// ---------------- atomic helpers (native, relaxed, device scope) ----------------
__device__ __forceinline__ void atomic_fadd(float* p, float v) {
    __hip_atomic_fetch_add(p, v, __ATOMIC_RELAXED, __HIP_MEMORY_SCOPE_AGENT);
}
// float max via signed/unsigned int ordering trick (works with -inf init)
__device__ __forceinline__ void atomic_fmax(float* addr, float value) {
    if (value >= 0.0f) {
        __hip_atomic_fetch_max((int*)addr, __float_as_int(value),
                               __ATOMIC_RELAXED, __HIP_MEMORY_SCOPE_AGENT);
    } else {
        __hip_atomic_fetch_min((unsigned int*)addr, __float_as_uint(value),
                               __ATOMIC_RELAXED, __HIP_MEMORY_SCOPE_AGENT);
    }
}

__device__ __forceinline__ float gelu_exact(float x) {
    return 0.5f * x * (1.0f + erff(x * 0.70710678118654752440f));
}

// ---------------- WMMA f32 GEMM:  C = act(A[M,K] @ B[K,Nc] + bias) ----------------
// Block = 256 threads = 8 waves; block tile 128 rows x 64 cols.
// Each wave computes a 16x64 strip (four 16x16 tiles sharing one A fragment).
// B is staged per K-chunk (128x64 f32 = 32 KB) into LDS via async global->LDS.
// Fragment layouts per CDNA5 ISA 7.12.2:
//   A 16x4 : lanes 0-15 rows 0-15 with K={k,k+1}; lanes 16-31 same rows, K={k+2,k+3}
//   B 4x16 : lanes 0-15 cols, rows {k,k+1}; lanes 16-31 cols, rows {k+2,k+3}
//   C 16x16: VGPR v, lanes 0-15 -> row v, lanes 16-31 -> row v+8
__global__ void __launch_bounds__(256)
wmma_gemm_f32(const float* __restrict__ A, const float* __restrict__ B,
              const float* __restrict__ bias, float* __restrict__ C,
              int M, int K, int Nc, int act)
{
    __shared__ float Bs[KCHUNK * 64];     // 32 KB

    const int lane   = threadIdx.x & 31;
    const int wave   = threadIdx.x >> 5;
    const int laneLo = lane & 15;
    const int halfK  = (lane >> 4) << 1;   // 0 or 2 : K sub-offset for A/B frags

    const int rowBase = (blockIdx.y * 8 + wave) * 16;
    const int colBase = blockIdx.x * 64;

    // Clamp the A row: out-of-range tile rows compute garbage that is never
    // stored, keeping loads unconditional (no EXEC divergence in the hot loop).
    int rowA = rowBase + laneLo;
    if (rowA >= M) rowA = M - 1;
    const float* Arow = A + (size_t)rowA * K;

    // cooperative stage mapping: 256 threads, each moves 16B per step
    const int stC = (threadIdx.x & 15) * 4;   // col offset within 64 (float4)
    const int stK = threadIdx.x >> 4;         // 0..15 row offset within chunk

    v8f acc0 = {}, acc1 = {}, acc2 = {}, acc3 = {};

    for (int k0 = 0; k0 < K; k0 += KCHUNK) {
        // ---- stage B[k0 .. k0+KCHUNK) x [colBase .. colBase+64) into LDS ----
        #pragma unroll
        for (int p = 0; p < KCHUNK; p += 16) {
            const int kk = stK + p;
            stage16_to_lds(&Bs[kk * 64 + stC],
                           B + (size_t)(k0 + kk) * Nc + colBase + stC);
        }
        stage_wait();
        __syncthreads();

        const float* Ak = Arow + k0 + halfK;
        #pragma unroll 4
        for (int kk = 0; kk < KCHUNK; kk += 4) {
            v2f a;
            a.x = Ak[kk];
            a.y = Ak[kk + 1];
            const float* b_lo = &Bs[(kk + halfK) * 64 + laneLo];
            const float* b_hi = b_lo + 64;
            v2f b;
            b.x = b_lo[0];  b.y = b_hi[0];
            acc0 = __builtin_amdgcn_wmma_f32_16x16x4_f32(false, a, false, b,
                                                         (short)0, acc0, false, false);
            b.x = b_lo[16]; b.y = b_hi[16];
            acc1 = __builtin_amdgcn_wmma_f32_16x16x4_f32(false, a, false, b,
                                                         (short)0, acc1, false, false);
            b.x = b_lo[32]; b.y = b_hi[32];
            acc2 = __builtin_amdgcn_wmma_f32_16x16x4_f32(false, a, false, b,
                                                         (short)0, acc2, false, false);
            b.x = b_lo[48]; b.y = b_hi[48];
            acc3 = __builtin_amdgcn_wmma_f32_16x16x4_f32(false, a, false, b,
                                                         (short)0, acc3, false, false);
        }
        __syncthreads();
    }

    // ---- epilogue: bias + activation + guarded store ----
    const int rowC = rowBase + ((lane >> 4) << 3);   // +0 (lanes 0-15) or +8
    v8f accs[4] = {acc0, acc1, acc2, acc3};
    #pragma unroll
    for (int t = 0; t < 4; ++t) {
        const int col = colBase + t * 16 + laneLo;
        const float bb = bias ? bias[col] : 0.0f;
        #pragma unroll
        for (int v = 0; v < 8; ++v) {
            const int r = rowC + v;
            if (r >= M) continue;
            float xv = accs[t][v] + bb;
            if (act == 1) xv = gelu_exact(xv);
            C[(size_t)r * Nc + col] = xv;
        }
    }
}

// ---------------- attention logits: al_src/al_dst [N,H] ----------------
// one wave per node; lane l covers dims [8l, 8l+8) -> head = l>>3
__global__ void __launch_bounds__(256)
attn_logits_kernel(const float* __restrict__ h, const float* __restrict__ a_src,
                   const float* __restrict__ a_dst, float* __restrict__ als,
                   float* __restrict__ ald, int N)
{
    const int node = (int)(((long long)blockIdx.x * blockDim.x + threadIdx.x) >> 5);
    const int lane = threadIdx.x & 31;
    if (node >= N) return;

    const int head      = lane >> 3;
    const int dimInHead = (lane & 7) * 8;

    const float* hp = h + (size_t)node * DIMC + lane * 8;
    const float* as = a_src + head * HEADDIM + dimInHead;
    const float* ad = a_dst + head * HEADDIM + dimInHead;

    float s = 0.0f, d = 0.0f;
    #pragma unroll
    for (int j = 0; j < 8; ++j) {
        const float v = hp[j];
        s += v * as[j];
        d += v * ad[j];
    }
    // reduce across the 8 lanes of this head group
    for (int m = 4; m; m >>= 1) {
        s += __shfl_xor(s, m, 8);
        d += __shfl_xor(d, m, 8);
    }
    if ((lane & 7) == 0) {
        als[node * HEADS + head] = s;
        ald[node * HEADS + head] = d;
    }
}

// ---------------- fill ----------------
__global__ void fill_f32(float* __restrict__ p, float v, int n)
{
    const int i = blockIdx.x * blockDim.x + threadIdx.x;
    if (i < n) p[i] = v;
}

// ---------------- edge pass 1: segment max of leaky_relu logits ----------------
__global__ void __launch_bounds__(256)
edge_max_kernel(const long long* __restrict__ src, const long long* __restrict__ dst,
                const float* __restrict__ als, const float* __restrict__ ald,
                float* __restrict__ emax, long long E, int N)
{
    const long long i = (long long)blockIdx.x * blockDim.x + threadIdx.x;
    const long long total = E + N;
    if (i >= total) return;

    int s, d;
    if (i < E) { s = (int)src[i]; d = (int)dst[i]; }
    else       { s = d = (int)(i - E); }

    const float4 as = *(const float4*)(als + (size_t)s * HEADS);
    const float4 ad = *(const float4*)(ald + (size_t)d * HEADS);
    float e[4] = { as.x + ad.x, as.y + ad.y, as.z + ad.z, as.w + ad.w };
    #pragma unroll
    for (int hh = 0; hh < 4; ++hh) {
        float v = e[hh];
        v = (v > 0.0f) ? v : v * NEG_SLOPE;
        atomic_fmax(&emax[(size_t)d * HEADS + hh], v);
    }
}

// ---------------- edge pass 2: exp and segment sum ----------------
__global__ void __launch_bounds__(256)
edge_expsum_kernel(const long long* __restrict__ src, const long long* __restrict__ dst,
                   const float* __restrict__ als, const float* __restrict__ ald,
                   const float* __restrict__ emax, float* __restrict__ denom,
                   float* __restrict__ exv, long long E, int N)
{
    const long long i = (long long)blockIdx.x * blockDim.x + threadIdx.x;
    const long long total = E + N;
    if (i >= total) return;

    int s, d;
    if (i < E) { s = (int)src[i]; d = (int)dst[i]; }
    else       { s = d = (int)(i - E); }

    const float4 as = *(const float4*)(als + (size_t)s * HEADS);
    const float4 ad = *(const float4*)(ald + (size_t)d * HEADS);
    const float4 em = *(const float4*)(emax + (size_t)d * HEADS);
    float e[4]  = { as.x + ad.x, as.y + ad.y, as.z + ad.z, as.w + ad.w };
    float mx[4] = { em.x, em.y, em.z, em.w };
    float4 exq;
    float* exa = (float*)&exq;
    #pragma unroll
    for (int hh = 0; hh < 4; ++hh) {
        float v = e[hh];
        v = (v > 0.0f) ? v : v * NEG_SLOPE;
        const float ex = expf(v - mx[hh]);
        exa[hh] = ex;
        atomic_fadd(&denom[(size_t)d * HEADS + hh], ex);
    }
    *(float4*)(exv + i * HEADS) = exq;
}

// ---------------- edge pass 3: out[dst] += alpha * h[src] ----------------
// one wave per edge; lane l handles dims [8l, 8l+8) (head = l>>3)
__global__ void __launch_bounds__(256)
edge_aggregate_kernel(const long long* __restrict__ src, const long long* __restrict__ dst,
                      const float* __restrict__ h, const float* __restrict__ exv,
                      const float* __restrict__ denom, float* __restrict__ out,
                      long long E, int N)
{
    const long long wv = ((long long)blockIdx.x * blockDim.x + threadIdx.x) >> 5;
    const int lane = threadIdx.x & 31;
    const long long total = E + N;
    if (wv >= total) return;

    int s, d;
    if (wv < E) { s = (int)src[wv]; d = (int)dst[wv]; }
    else        { s = d = (int)(wv - E); }

    const int   head  = lane >> 3;
    const float alpha = exv[wv * HEADS + head] / denom[(size_t)d * HEADS + head];

    const float4* hp = (const float4*)(h + (size_t)s * DIMC + lane * 8);
    const float4 v0 = hp[0];
    const float4 v1 = hp[1];

    float* op = out + (size_t)d * DIMC + lane * 8;
    atomic_fadd(op + 0, alpha * v0.x);
    atomic_fadd(op + 1, alpha * v0.y);
    atomic_fadd(op + 2, alpha * v0.z);
    atomic_fadd(op + 3, alpha * v0.w);
    atomic_fadd(op + 4, alpha * v1.x);
    atomic_fadd(op + 5, alpha * v1.y);
    atomic_fadd(op + 6, alpha * v1.z);
    atomic_fadd(op + 7, alpha * v1.w);
}

// ---------------- fused residual + LayerNorm (one wave per 256-wide row) ----------------
// out = LN(a + res (+ colbias)) * g + beta
__global__ void __launch_bounds__(256)
layernorm_res_kernel(const float* __restrict__ a, const float* __restrict__ colbias,
                     const float* __restrict__ res, const float* __restrict__ g,
                     const float* __restrict__ beta, float* __restrict__ out, int M)
{
    const int row  = (int)(((long long)blockIdx.x * blockDim.x + threadIdx.x) >> 5);
    const int lane = threadIdx.x & 31;
    if (row >= M) return;

    const size_t base = (size_t)row * DIMC;
    float vals[8];
    float sum = 0.0f, sq = 0.0f;
    #pragma unroll
    for (int j = 0; j < 8; ++j) {
        const int idx = j * 32 + lane;
        float v = a[base + idx] + res[base + idx];
        if (colbias) v += colbias[idx];
        vals[j] = v;
        sum += v;
        sq  += v * v;
    }
    for (int m = 16; m; m >>= 1) {
        sum += __shfl_xor(sum, m, 32);
        sq  += __shfl_xor(sq,  m, 32);
    }
    const float mean = sum * (1.0f / DIMC);
    const float var  = sq * (1.0f / DIMC) - mean * mean;
    const float rstd = rsqrtf(var + LN_EPS);
    #pragma unroll
    for (int j = 0; j < 8; ++j) {
        const int idx = j * 32 + lane;
        out[base + idx] = (vals[j] - mean) * rstd * g[idx] + beta[idx];
    }
}

// ---------------- host-side orchestration ----------------
extern "C" void kernel_launch(void* const* d_in, const int* in_sizes, int n_in,
                              void* d_out, int out_size, void* d_ws, size_t ws_size,
                              hipStream_t stream)
{
    const float*     x     = (const float*)d_in[0];
    const long long* eidx  = (const long long*)d_in[1];
    const float*     W     = (const float*)d_in[2];
    const float*     a_src = (const float*)d_in[3];
    const float*     a_dst = (const float*)d_in[4];
    const float*     gat_b = (const float*)d_in[5];
    const float*     ln1_g = (const float*)d_in[6];
    const float*     ln1_b = (const float*)d_in[7];
    const float*     W1    = (const float*)d_in[8];
    const float*     b1    = (const float*)d_in[9];
    const float*     W2    = (const float*)d_in[10];
    const float*     b2    = (const float*)d_in[11];
    const float*     ln2_g = (const float*)d_in[12];
    const float*     ln2_b = (const float*)d_in[13];

    const int       N   = in_sizes[0] / DIMC;
    const long long E   = (long long)in_sizes[1] / 2;
    const long long EN  = E + N;
    const long long* srcp = eidx;
    const long long* dstp = eidx + E;

    // workspace carve-out (256B aligned regions), with aliasing:
    //   hbuf: h, then reused as y1   |   obuf: gat-out, then reused as f
    char* w = (char*)d_ws;
    auto carve = [&](size_t bytes) -> float* {
        float* p = (float*)w;
        w += (bytes + 255) & ~(size_t)255;
        return p;
    };
    float* hbuf  = carve((size_t)N * DIMC * sizeof(float));       // h  -> y1
    float* obuf  = carve((size_t)N * DIMC * sizeof(float));       // out-> f
    float* gbuf  = carve((size_t)N * 2 * DIMC * sizeof(float));   // gelu intermediate
    float* exbuf = carve((size_t)EN * HEADS * sizeof(float));     // per-edge exp
    float* als   = carve((size_t)N * HEADS * sizeof(float));
    float* ald   = carve((size_t)N * HEADS * sizeof(float));
    float* emax  = carve((size_t)N * HEADS * sizeof(float));
    float* denom = carve((size_t)N * HEADS * sizeof(float));
    (void)ws_size; (void)n_in; (void)out_size;

    // 1) h = x @ W
    {
        dim3 grid(DIMC / 64, (N + 127) / 128);
        wmma_gemm_f32<<<grid, 256, 0, stream>>>(x, W, nullptr, hbuf, N, DIMC, DIMC, 0);
    }
    // 2) attention logits
    {
        const int blocks = (N + 7) / 8;   // 8 waves/block, 1 wave/node
        attn_logits_kernel<<<blocks, 256, 0, stream>>>(hbuf, a_src, a_dst, als, ald, N);
    }
    // 3) init accumulators
    {
        const int nh = N * HEADS;
        fill_f32<<<(nh + 255) / 256, 256, 0, stream>>>(emax, -__builtin_inff(), nh);
        fill_f32<<<(nh + 255) / 256, 256, 0, stream>>>(denom, 0.0f, nh);
        const int nd = N * DIMC;
        fill_f32<<<(nd + 255) / 256, 256, 0, stream>>>(obuf, 0.0f, nd);
    }
    // 4) segment max over edges (+self loops)
    {
        const int blocks = (int)((EN + 255) / 256);
        edge_max_kernel<<<blocks, 256, 0, stream>>>(srcp, dstp, als, ald, emax, E, N);
    }
    // 5) exp + segment sum
    {
        const int blocks = (int)((EN + 255) / 256);
        edge_expsum_kernel<<<blocks, 256, 0, stream>>>(srcp, dstp, als, ald, emax,
                                                       denom, exbuf, E, N);
    }
    // 6) weighted aggregation (wave per edge)
    {
        const long long threads = EN * 32;
        const int blocks = (int)((threads + 255) / 256);
        edge_aggregate_kernel<<<blocks, 256, 0, stream>>>(srcp, dstp, hbuf, exbuf,
                                                          denom, obuf, E, N);
    }
    // 7) y1 = LN(out + gat_b + x)   (y1 overwrites hbuf; h fully consumed)
    float* y1 = hbuf;
    {
        const int blocks = (N + 7) / 8;
        layernorm_res_kernel<<<blocks, 256, 0, stream>>>(obuf, gat_b, x, ln1_g, ln1_b, y1, N);
    }
    // 8) g = gelu(y1 @ W1 + b1)
    {
        dim3 grid((2 * DIMC) / 64, (N + 127) / 128);
        wmma_gemm_f32<<<grid, 256, 0, stream>>>(y1, W1, b1, gbuf, N, DIMC, 2 * DIMC, 1);
    }
    // 9) f = g @ W2 + b2            (f overwrites obuf; gat-out fully consumed)
    float* fbuf = obuf;
    {
        dim3 grid(DIMC / 64, (N + 127) / 128);
        wmma_gemm_f32<<<grid, 256, 0, stream>>>(gbuf, W2, b2, fbuf, N, 2 * DIMC, DIMC, 0);
    }
    // 10) d_out = LN(f + y1)
    {
        const int blocks = (N + 7) / 8;
        layernorm_res_kernel<<<blocks, 256, 0, stream>>>(fbuf, nullptr, y1, ln2_g, ln2_b,
                                                         (float*)d_out, N);
    }
}